// knnLoss_26371099197710
// MI455X (gfx1250) — compile-verified
//
#include <hip/hip_runtime.h>

typedef __attribute__((ext_vector_type(2))) float v2f;
typedef __attribute__((ext_vector_type(8))) float v8f;

#define B_BATCH   2
#define NPTS      16384      // 32*512 downsampled points
#define NTILES    (NPTS/16)  // 1024 column tiles
#define SRC_H     64
#define SRC_W     2048

// Branchless sorted-min3 insert: a0<=a1<=a2 stay the 3 smallest seen.
__device__ __forceinline__ void min3_insert(float v, float& a0, float& a1, float& a2) {
    float t0 = fminf(v, a0);
    float t1 = fmaxf(v, a0);
    float u1 = fminf(t1, a1);
    float u2 = fmaxf(t1, a1);
    a0 = t0;
    a1 = u1;
    a2 = fminf(u2, a2);
}

__global__ void knn_init(float* sum_d, unsigned* cnt) {
    if (threadIdx.x < B_BATCH) {
        sum_d[threadIdx.x] = 0.0f;
        cnt[threadIdx.x]   = 0u;
    }
}

// Pack strided downsample into contiguous float4 {x,y,z,|p|^2}; count valid src.
__global__ void knn_pack(const float* __restrict__ src,   // [B,64,2048,3]
                         const float* __restrict__ tgt,   // [B,3,64,2048]
                         float4* __restrict__ srcp,       // [B,NPTS]
                         float4* __restrict__ tgtp,       // [B,NPTS]
                         unsigned* __restrict__ cnt) {
    int gid  = blockIdx.x * blockDim.x + threadIdx.x;   // [0, 2*B*NPTS)
    int p    = gid & (NPTS - 1);
    int rest = gid >> 14;
    int b    = rest >> 1;
    int kind = rest & 1;                                // 0=src, 1=tgt
    int oh = p >> 9, ow = p & 511;                      // 32 x 512
    float x, y, z;
    if (kind == 0) {
        long idx = (((long)b * SRC_H + oh * 2) * SRC_W + ow * 4) * 3;
        x = src[idx]; y = src[idx + 1]; z = src[idx + 2];
    } else {
        long base = ((long)b * 3 * SRC_H + oh * 2) * SRC_W + (long)ow * 4;
        x = tgt[base];
        y = tgt[base + (long)SRC_H * SRC_W];
        z = tgt[base + 2l * SRC_H * SRC_W];
    }
    float n = x * x + y * y + z * z;                    // 0 iff all-zero point
    float4 v = make_float4(x, y, z, n);
    if (kind == 0) {
        srcp[b * NPTS + p] = v;
        unsigned long long bal = __ballot(n != 0.0f);   // warp-uniform kind
        if ((threadIdx.x & 31) == 0)
            atomicAdd(&cnt[b], (unsigned)__popcll(bal));
    } else {
        tgtp[b * NPTS + p] = v;
    }
}

// One wave32 per 16 source rows: WMMA rank-4 cross term + streaming min-3.
__global__ __launch_bounds__(32) void knn_tile(const float4* __restrict__ srcp,
                                               const float4* __restrict__ tgtp,
                                               float* __restrict__ sum_d) {
    int wid  = blockIdx.x;
    int b    = wid / NTILES;
    int tile = wid % NTILES;
    int l    = threadIdx.x;
    int m    = l & 15;
    bool hi  = (l >= 16);

    const float4* sb = srcp + b * NPTS;
    const float4* tb = tgtp + b * NPTS;

    // A fragment (16x4 f32, 2 VGPRs): row M = [-2x, -2y, -2z, |s|^2]
    // lanes 0-15: K=0,1 ; lanes 16-31: K=2,3
    float4 sp = sb[tile * 16 + m];
    v2f A;
    A.x = hi ? (-2.0f * sp.z) : (-2.0f * sp.x);
    A.y = hi ? (sp.w)         : (-2.0f * sp.y);
    float spw = sp.w;

    float m0[8], m1[8], m2[8];
#pragma unroll
    for (int r = 0; r < 8; ++r) { m0[r] = 1e30f; m1[r] = 1e30f; m2[r] = 1e30f; }

    for (int j = 0; j < NTILES; ++j) {
        float4 tp = tb[j * 16 + m];
        // B fragment (4x16 f32, 2 VGPRs): col N = [x, y, z, 1]
        v2f Bf;
        Bf.x = hi ? tp.z : tp.x;
        Bf.y = hi ? 1.0f : tp.y;
        float tn   = tp.w;               // |t|^2 for column N = l&15
        bool  tval = (tn != 0.0f);

        v8f c = {};
        // D = A x B + 0 = |s_M|^2 - 2 s_M . t_N   (tile 16x16)
        c = __builtin_amdgcn_wmma_f32_16x16x4_f32(
                /*neg_a=*/false, A, /*neg_b=*/false, Bf,
                /*c_mod=*/(short)0, c, /*reuse_a=*/false, /*reuse_b=*/false);

#pragma unroll
        for (int r = 0; r < 8; ++r) {
            float d2 = c[r] + tn;            // + |t|^2
            d2 = fmaxf(d2, 0.0f);
            d2 = tval ? d2 : 1e20f;          // sqrt(1e20) == reference 1e10 mask
            min3_insert(d2, m0[r], m1[r], m2[r]);
        }
    }

    // Merge min-3 across the 16 lanes of each half (xor masks stay in-half).
#pragma unroll
    for (int mask = 1; mask <= 8; mask <<= 1) {
#pragma unroll
        for (int r = 0; r < 8; ++r) {
            float o0 = __shfl_xor(m0[r], mask, 32);
            float o1 = __shfl_xor(m1[r], mask, 32);
            float o2 = __shfl_xor(m2[r], mask, 32);
            min3_insert(o0, m0[r], m1[r], m2[r]);
            min3_insert(o1, m0[r], m1[r], m2[r]);
            min3_insert(o2, m0[r], m1[r], m2[r]);
        }
    }

    // Row R = r + 8*hi; its validity (|s|^2) lives in lane (r + 8*hi).
    float local = 0.0f;
#pragma unroll
    for (int r = 0; r < 8; ++r) {
        float w = __shfl(spw, r + (hi ? 8 : 0), 32);   // all lanes participate
        float s = sqrtf(m0[r]) + sqrtf(m1[r]) + sqrtf(m2[r]);
        local += (w != 0.0f) ? s : 0.0f;
    }
    if (m == 0)                                        // lanes 0 and 16
        atomicAdd(&sum_d[b], local);
}

__global__ void knn_finalize(const float* __restrict__ sum_d,
                             const unsigned* __restrict__ cnt,
                             float* __restrict__ out) {
    if (threadIdx.x == 0 && blockIdx.x == 0) {
        float acc = 0.0f;
        for (int b = 0; b < B_BATCH; ++b) {
            float den = 3.0f * fmaxf((float)cnt[b], 1.0f);
            acc += sum_d[b] / den;
        }
        out[0] = acc / (float)B_BATCH;
    }
}

extern "C" void kernel_launch(void* const* d_in, const int* in_sizes, int n_in,
                              void* d_out, int out_size, void* d_ws, size_t ws_size,
                              hipStream_t stream) {
    const float* src = (const float*)d_in[0];   // [2,64,2048,3]
    const float* tgt = (const float*)d_in[1];   // [2,3,64,2048]
    float* out = (float*)d_out;

    // Workspace layout: src_pack | tgt_pack | sum_d[2] | cnt[2]
    float4*   srcp  = (float4*)d_ws;                       // B*NPTS float4
    float4*   tgtp  = srcp + (size_t)B_BATCH * NPTS;       // B*NPTS float4
    float*    sum_d = (float*)(tgtp + (size_t)B_BATCH * NPTS);
    unsigned* cnt   = (unsigned*)(sum_d + B_BATCH);

    knn_init<<<1, 32, 0, stream>>>(sum_d, cnt);

    int pack_threads = 2 * B_BATCH * NPTS;                 // 65536
    knn_pack<<<pack_threads / 256, 256, 0, stream>>>(src, tgt, srcp, tgtp, cnt);

    knn_tile<<<B_BATCH * NTILES, 32, 0, stream>>>(srcp, tgtp, sum_d);

    knn_finalize<<<1, 32, 0, stream>>>(sum_d, cnt, out);
}